// LocSmoothL1Loss_65635690217876
// MI455X (gfx1250) — compile-verified
//
#include <hip/hip_runtime.h>

#define TOPK_N 16
#define NTHREADS 256
#define BLOCKS_PER_BATCH 32
#define BATCH 32
#define IMG_H 2048
#define IMG_W 2048
#define ELEMS_PER_BATCH (IMG_H * IMG_W)                 // 4194304
#define CHUNK (ELEMS_PER_BATCH / BLOCKS_PER_BATCH)      // 131072 elems per block
#define ITERS (CHUNK / (NTHREADS * 4))                  // 128 float4 iters per thread
#define PIPE 4
#define NEG_BIG (-3.4028234663852886e+38f)

#if defined(__HIP_DEVICE_COMPILE__)
  #define ON_DEVICE 1
#else
  #define ON_DEVICE 0
#endif

typedef int v4i_t __attribute__((vector_size(16)));

// ---------------- CDNA5 split-counter waits ----------------
template <int N>
__device__ __forceinline__ void s_wait_asynccnt_le() {
#if ON_DEVICE
#if __has_builtin(__builtin_amdgcn_s_wait_asynccnt)
  __builtin_amdgcn_s_wait_asynccnt(N);
#else
  asm volatile("s_wait_asynccnt %0" ::"i"(N) : "memory");
#endif
#endif
}

template <int N>
__device__ __forceinline__ void s_wait_dscnt_le() {
#if ON_DEVICE
#if __has_builtin(__builtin_amdgcn_s_wait_dscnt)
  __builtin_amdgcn_s_wait_dscnt(N);
#else
  asm volatile("s_wait_dscnt %0" ::"i"(N) : "memory");
#endif
#endif
}

// ---------------- async global -> LDS 16B/lane copy ----------------
__device__ __forceinline__ void async_copy_b128(const void* gp, void* lp) {
#if ON_DEVICE
#if __has_builtin(__builtin_amdgcn_global_load_async_to_lds_b128)
  __builtin_amdgcn_global_load_async_to_lds_b128(
      (__attribute__((address_space(1))) v4i_t*)gp,
      (__attribute__((address_space(3))) v4i_t*)lp, 0, 0);
#else
  unsigned lds_off =
      (unsigned)(unsigned long long)(__attribute__((address_space(3))) void*)lp;
  unsigned long long ga = (unsigned long long)gp;
  // VDST = LDS byte-address VGPR, VADDR = 64-bit global address, GV mode
  asm volatile("global_load_async_to_lds_b128 %0, %1, off"
               ::"v"(lds_off), "v"(ga)
               : "memory");
#endif
#endif
}

// ---------------- branchless register-resident top-16 insert ----------------
// v[] sorted descending; all indices constant after unroll -> stays in VGPRs.
__device__ __forceinline__ void topk_insert(float (&v)[TOPK_N], int (&ix)[TOPK_N],
                                            float x, int gi) {
  if (x <= v[TOPK_N - 1]) return;  // common path: 1 compare per element
#pragma unroll
  for (int j = TOPK_N - 1; j >= 1; --j) {
    bool a = x > v[j];
    bool b = x > v[j - 1];
    float nv = a ? (b ? v[j - 1] : x) : v[j];
    int ni = a ? (b ? ix[j - 1] : gi) : ix[j];
    v[j] = nv;
    ix[j] = ni;
  }
  if (x > v[0]) { v[0] = x; ix[0] = gi; }
}

// Merge two sorted-descending 16-lists held in LDS into list[dst].
__device__ __forceinline__ void lds_merge_pair(float* sv, int* si, int dst, int src) {
  float tv[TOPK_N];
  int ti[TOPK_N];
  int i = 0, j = 0;
#pragma unroll
  for (int k = 0; k < TOPK_N; ++k) {
    float va = sv[dst * TOPK_N + i];
    float vb = sv[src * TOPK_N + j];
    bool ta = va >= vb;
    tv[k] = ta ? va : vb;
    ti[k] = ta ? si[dst * TOPK_N + i] : si[src * TOPK_N + j];
    if (ta) ++i; else ++j;
  }
#pragma unroll
  for (int k = 0; k < TOPK_N; ++k) {
    sv[dst * TOPK_N + k] = tv[k];
    si[dst * TOPK_N + k] = ti[k];
  }
}

__device__ __forceinline__ void block_topk_merge(float* sv, int* si, int tid) {
  for (int stride = 1; stride < NTHREADS; stride <<= 1) {
    __syncthreads();
    if ((tid & (2 * stride - 1)) == 0) lds_merge_pair(sv, si, tid, tid + stride);
  }
  __syncthreads();
}

__device__ __forceinline__ float smooth_l1(float p, float t) {
  float d = p - t;
  float ad = fabsf(d);
  return (ad < 1.0f) ? (0.5f * d * d) : (ad - 0.5f);
}

// =================== Phase 1: per-block top-16 of 131072 elems ===================
__global__ __launch_bounds__(NTHREADS) void phase1_topk(const float* __restrict__ in,
                                                        float* __restrict__ out_vals,
                                                        int* __restrict__ out_idx) {
  __shared__ float sv[NTHREADS * TOPK_N];
  __shared__ int si[NTHREADS * TOPK_N];
  __shared__ float4 stage[PIPE][NTHREADS];

  const int tid = threadIdx.x;
  const int chunk = blockIdx.x;
  const int b = blockIdx.y;

  const float4* __restrict__ src4 =
      (const float4*)(in + (size_t)b * ELEMS_PER_BATCH + (size_t)chunk * CHUNK) + tid;

  float v[TOPK_N];
  int ix[TOPK_N];
#pragma unroll
  for (int k = 0; k < TOPK_N; ++k) { v[k] = NEG_BIG; ix[k] = 0; }

  const int gbase = chunk * CHUNK + tid * 4;  // element index within this batch

  auto process = [&](const float4& d, int k) {
    const int gi = gbase + k * (NTHREADS * 4);
    topk_insert(v, ix, d.x, gi + 0);
    topk_insert(v, ix, d.y, gi + 1);
    topk_insert(v, ix, d.z, gi + 2);
    topk_insert(v, ix, d.w, gi + 3);
  };

  // ---- 4-deep async global->LDS pipeline (ASYNCcnt-paced) ----
#pragma unroll
  for (int p = 0; p < PIPE; ++p)
    async_copy_b128(src4 + (size_t)p * NTHREADS, &stage[p][tid]);

  for (int k = 0; k < ITERS - PIPE; ++k) {
    const int slot = k & (PIPE - 1);
    s_wait_asynccnt_le<PIPE - 1>();           // oldest async copy has landed
    float4 d = stage[slot][tid];              // each lane reads only its own slot
    s_wait_dscnt_le<0>();                     // slot free before re-issuing into it
    async_copy_b128(src4 + (size_t)(k + PIPE) * NTHREADS, &stage[slot][tid]);
    process(d, k);
  }
  s_wait_asynccnt_le<0>();                    // drain tail
  for (int k = ITERS - PIPE; k < ITERS; ++k) {
    float4 d = stage[k & (PIPE - 1)][tid];
    process(d, k);
  }

  // ---- block-level tree merge in LDS ----
#pragma unroll
  for (int k = 0; k < TOPK_N; ++k) {
    sv[tid * TOPK_N + k] = v[k];
    si[tid * TOPK_N + k] = ix[k];
  }
  block_topk_merge(sv, si, tid);

  if (tid < TOPK_N) {
    int o = (b * BLOCKS_PER_BATCH + chunk) * TOPK_N + tid;
    out_vals[o] = sv[tid];
    out_idx[o] = si[tid];
  }
}

// =================== Phase 2: per-batch final top-16 + partial loss ===================
__global__ __launch_bounds__(NTHREADS) void phase2_final(const float* __restrict__ cand_vals,
                                                         const int* __restrict__ cand_idx,
                                                         const float* __restrict__ center_rate,
                                                         float* __restrict__ partial) {
  __shared__ float sv[NTHREADS * TOPK_N];
  __shared__ int si[NTHREADS * TOPK_N];
  const int tid = threadIdx.x;
  const int b = blockIdx.x;
  const int NC = BLOCKS_PER_BATCH * TOPK_N;  // 512 candidates per batch

  float v[TOPK_N];
  int ix[TOPK_N];
#pragma unroll
  for (int k = 0; k < TOPK_N; ++k) { v[k] = NEG_BIG; ix[k] = 0; }

  for (int c = tid; c < NC; c += NTHREADS)
    topk_insert(v, ix, cand_vals[b * NC + c], cand_idx[b * NC + c]);

#pragma unroll
  for (int k = 0; k < TOPK_N; ++k) {
    sv[tid * TOPK_N + k] = v[k];
    si[tid * TOPK_N + k] = ix[k];
  }
  block_topk_merge(sv, si, tid);

  if (tid == 0) {
    float t0 = center_rate[b * 2 + 0];
    float t1 = center_rate[b * 2 + 1];
    float s = 0.0f;
#pragma unroll
    for (int k = 0; k < TOPK_N; ++k) {
      int id = si[k];
      float r = (float)(id / IMG_W);
      float c = (float)(id % IMG_W);
      s += smooth_l1(r / (float)(IMG_H - 1), t0);
      s += smooth_l1(c / (float)(IMG_W - 1), t1);
    }
    partial[b] = s;
  }
}

// =================== Phase 3: reduce partials -> scalar loss ===================
__global__ void phase3_reduce(const float* __restrict__ partial, float* __restrict__ out) {
  if (threadIdx.x == 0) {
    float s = 0.0f;
#pragma unroll
    for (int b = 0; b < BATCH; ++b) s += partial[b];
    out[0] = s * (1.0f / (float)(BATCH * TOPK_N * 2));  // mean * WEIGHT_RATE(=1)
  }
}

extern "C" void kernel_launch(void* const* d_in, const int* in_sizes, int n_in,
                              void* d_out, int out_size, void* d_ws, size_t ws_size,
                              hipStream_t stream) {
  (void)in_sizes; (void)n_in; (void)out_size; (void)ws_size;
  const float* cls = (const float*)d_in[0];          // (32,1,2048,2048) fp32
  const float* center = (const float*)d_in[1];       // (32,2) fp32

  const int ncand = BATCH * BLOCKS_PER_BATCH * TOPK_N;  // 16384
  float* ws_vals = (float*)d_ws;
  int* ws_idx = (int*)(ws_vals + ncand);
  float* ws_part = (float*)(ws_idx + ncand);

  dim3 g1(BLOCKS_PER_BATCH, BATCH);
  phase1_topk<<<g1, NTHREADS, 0, stream>>>(cls, ws_vals, ws_idx);
  phase2_final<<<BATCH, NTHREADS, 0, stream>>>(ws_vals, ws_idx, center, ws_part);
  phase3_reduce<<<1, 32, 0, stream>>>(ws_part, (float*)d_out);
}